// GraphGATNet_37168646980026
// MI455X (gfx1250) — compile-verified
//
#include <hip/hip_runtime.h>
#include <hip/hip_bf16.h>
#include <math.h>

#define N_NODES 50000
#define N_EDGES 800000
#define E_TOT   (N_EDGES + N_NODES)   // edges + self loops
#define HEADS 8
#define HID 16
#define OUT_C 128
#define NCH 128                       // 128 channels at every stage
#define NEG_SLOPE 0.2f
#define GAT_EPS 1e-16f

typedef __attribute__((ext_vector_type(2))) float v2f;
typedef __attribute__((ext_vector_type(8))) float v8f;

#define GRID1(n) ((int)(((long)(n) + 255) / 256))

// ---------------------------------------------------------------- utilities

__global__ void k_fill(float* __restrict__ p, float val, long count) {
    long i = (long)blockIdx.x * blockDim.x + threadIdx.x;
    if (i < count) p[i] = val;
}

__device__ __forceinline__ void atomicMaxF(float* addr, float v) {
    // monotone int/uint mapping of IEEE floats (init value is -inf)
    if (v >= 0.0f) atomicMax((int*)addr, __float_as_int(v));
    else           atomicMin((unsigned int*)addr, (unsigned int)__float_as_int(v));
}

__device__ __forceinline__ float leaky(float v) {
    return v > 0.0f ? v : NEG_SLOPE * v;
}

// ---------------------------------------------------------------- WMMA GEMM
// H[N,128] = X[N,128] @ W[128,128], fp32 via V_WMMA_F32_16X16X4_F32.
// One wave -> one 16x16 tile; 8 waves/block -> 16x128 row strip.
__global__ __launch_bounds__(256) void k_gemm_wmma(
    const float* __restrict__ X, const float* __restrict__ W,
    float* __restrict__ Hout)
{
    const int lane = threadIdx.x & 31;
    const int wave = threadIdx.x >> 5;
    const int half = lane >> 4;     // 0: lanes 0-15, 1: lanes 16-31
    const int l16  = lane & 15;
    const long row0 = (long)blockIdx.x * 16;
    const int  col0 = wave * 16;

    const float* xrow = X + (row0 + l16) * NCH;   // A: M = l16 for both halves
    v8f c = {};

    #pragma unroll 4
    for (int k = 0; k < 128; k += 4) {
        const int ka = k + half * 2;              // K pair handled by this half
        // A 16x4: v0 = {K=k | K=k+2}, v1 = {K=k+1 | K=k+3}
        v2f a;
        a.x = xrow[ka + 0];
        a.y = xrow[ka + 1];
        // B 4x16: v0 = rows {k, k+2}, v1 = rows {k+1, k+3}, cols on lanes
        v2f b;
        b.x = W[(ka + 0) * NCH + col0 + l16];
        b.y = W[(ka + 1) * NCH + col0 + l16];
        c = __builtin_amdgcn_wmma_f32_16x16x4_f32(
                false, a, false, b, (short)0, c, false, false);
    }
    // C/D: VGPR r -> rows {r, r+8}, cols on lanes
    float* out = Hout + (row0 + half * 8) * NCH + col0 + l16;
    #pragma unroll
    for (int r = 0; r < 8; ++r) out[(long)r * NCH] = c[r];
}

// ------------------------------------------------- per-node attention coeffs
template<int H, int C>
__global__ void k_attn(const float* __restrict__ Hm,
                       const float* __restrict__ att_s,
                       const float* __restrict__ att_d,
                       float* __restrict__ a_s, float* __restrict__ a_d)
{
    long idx = (long)blockIdx.x * blockDim.x + threadIdx.x;
    if (idx >= (long)N_NODES * H) return;
    const int n = (int)(idx / H), h = (int)(idx % H);
    const float* hp = Hm + (long)n * (H * C) + h * C;
    const float* as = att_s + h * C;
    const float* ad = att_d + h * C;
    float ss = 0.0f, sd = 0.0f;
    #pragma unroll 8
    for (int ch = 0; ch < C; ++ch) { float v = hp[ch]; ss += v * as[ch]; sd += v * ad[ch]; }
    a_s[idx] = ss;
    a_d[idx] = sd;
}

// ------------------------------------------------------ edge softmax passes
template<int H>
__device__ __forceinline__ void edge_decode(long idx, const int* esrc,
                                            const int* edst, int& s, int& d, int& h)
{
    int e = (int)(idx / H);
    h = (int)(idx % H);
    if (e < N_EDGES) { s = esrc[e]; d = edst[e]; } else { s = e - N_EDGES; d = s; }
}

template<int H>
__global__ void k_edge_max(const int* __restrict__ esrc, const int* __restrict__ edst,
                           const float* __restrict__ a_s, const float* __restrict__ a_d,
                           float* __restrict__ m)
{
    long idx = (long)blockIdx.x * blockDim.x + threadIdx.x;
    if (idx >= (long)E_TOT * H) return;
    int s, d, h; edge_decode<H>(idx, esrc, edst, s, d, h);
    float v = leaky(a_s[(long)s * H + h] + a_d[(long)d * H + h]);
    atomicMaxF(&m[(long)d * H + h], v);
}

template<int H>
__global__ void k_edge_sum(const int* __restrict__ esrc, const int* __restrict__ edst,
                           const float* __restrict__ a_s, const float* __restrict__ a_d,
                           const float* __restrict__ m, float* __restrict__ den)
{
    long idx = (long)blockIdx.x * blockDim.x + threadIdx.x;
    if (idx >= (long)E_TOT * H) return;
    int s, d, h; edge_decode<H>(idx, esrc, edst, s, d, h);
    float v = leaky(a_s[(long)s * H + h] + a_d[(long)d * H + h]);
    atomicAdd(&den[(long)d * H + h], __expf(v - m[(long)d * H + h]));
}

template<int H, int C>
__global__ void k_edge_scatter(const int* __restrict__ esrc, const int* __restrict__ edst,
                               const float* __restrict__ a_s, const float* __restrict__ a_d,
                               const float* __restrict__ m, const float* __restrict__ den,
                               const float* __restrict__ Hm, float* __restrict__ out)
{
    long idx = (long)blockIdx.x * blockDim.x + threadIdx.x;
    if (idx >= (long)E_TOT * H) return;
    int s, d, h; edge_decode<H>(idx, esrc, edst, s, d, h);
    float v = leaky(a_s[(long)s * H + h] + a_d[(long)d * H + h]);
    float alpha = __expf(v - m[(long)d * H + h]) / (den[(long)d * H + h] + GAT_EPS);
    const float4* hs = (const float4*)(Hm + (long)s * (H * C) + h * C);
    float* op = out + (long)d * (H * C) + h * C;
    #pragma unroll
    for (int c4 = 0; c4 < C / 4; ++c4) {
        float4 x = hs[c4];
        atomicAdd(op + c4 * 4 + 0, alpha * x.x);
        atomicAdd(op + c4 * 4 + 1, alpha * x.y);
        atomicAdd(op + c4 * 4 + 2, alpha * x.z);
        atomicAdd(op + c4 * 4 + 3, alpha * x.w);
    }
}

// ------------------------------------------------------------- bias (+ReLU)
__global__ void k_bias_act(float* __restrict__ buf, const float* __restrict__ bias,
                           int relu, long count)
{
    long i = (long)blockIdx.x * blockDim.x + threadIdx.x;
    if (i >= count) return;
    float v = buf[i] + bias[i & (NCH - 1)];
    if (relu) v = fmaxf(v, 0.0f);
    buf[i] = v;
}

// ----------------------------------------------------------------- launcher
extern "C" void kernel_launch(void* const* d_in, const int* in_sizes, int n_in,
                              void* d_out, int out_size, void* d_ws, size_t ws_size,
                              hipStream_t stream)
{
    const float* x        = (const float*)d_in[0];
    const int*   ei       = (const int*)  d_in[1];
    const float* W1       = (const float*)d_in[2];
    const float* att_src1 = (const float*)d_in[3];
    const float* att_dst1 = (const float*)d_in[4];
    const float* b1       = (const float*)d_in[5];
    const float* W2       = (const float*)d_in[6];
    const float* att_src2 = (const float*)d_in[7];
    const float* att_dst2 = (const float*)d_in[8];
    const float* b2       = (const float*)d_in[9];
    const int* esrc = ei;
    const int* edst = ei + N_EDGES;
    float* out = (float*)d_out;

    // workspace layout (floats)
    float* ws   = (float*)d_ws;
    float* h    = ws;                              // N*128 (h1, reused as h2)
    float* out1 = h    + (long)N_NODES * NCH;      // N*128
    float* a_s1 = out1 + (long)N_NODES * NCH;      // N*8
    float* a_d1 = a_s1 + (long)N_NODES * HEADS;
    float* m1   = a_d1 + (long)N_NODES * HEADS;
    float* den1 = m1   + (long)N_NODES * HEADS;
    float* a_s2 = den1 + (long)N_NODES * HEADS;    // N
    float* a_d2 = a_s2 + N_NODES;
    float* m2   = a_d2 + N_NODES;
    float* den2 = m2   + N_NODES;

    const int T = 256;
    const long nNC = (long)N_NODES * NCH;
    const long nE1 = (long)E_TOT * HEADS;
    const long nE2 = (long)E_TOT;

    // re-init accumulators every call (deterministic work, harness poisons once)
    k_fill<<<GRID1(nNC), T, 0, stream>>>(out1, 0.0f, nNC);
    k_fill<<<GRID1(nNC), T, 0, stream>>>(out,  0.0f, nNC);
    k_fill<<<GRID1(N_NODES * HEADS), T, 0, stream>>>(m1,  -INFINITY, N_NODES * HEADS);
    k_fill<<<GRID1(N_NODES * HEADS), T, 0, stream>>>(den1, 0.0f,     N_NODES * HEADS);
    k_fill<<<GRID1(N_NODES), T, 0, stream>>>(m2,  -INFINITY, N_NODES);
    k_fill<<<GRID1(N_NODES), T, 0, stream>>>(den2, 0.0f,     N_NODES);

    // ---------------- layer 1: 128 -> 8 heads x 16, concat ----------------
    k_gemm_wmma<<<N_NODES / 16, 256, 0, stream>>>(x, W1, h);
    k_attn<HEADS, HID><<<GRID1((long)N_NODES * HEADS), T, 0, stream>>>(
        h, att_src1, att_dst1, a_s1, a_d1);
    k_edge_max<HEADS><<<GRID1(nE1), T, 0, stream>>>(esrc, edst, a_s1, a_d1, m1);
    k_edge_sum<HEADS><<<GRID1(nE1), T, 0, stream>>>(esrc, edst, a_s1, a_d1, m1, den1);
    k_edge_scatter<HEADS, HID><<<GRID1(nE1), T, 0, stream>>>(
        esrc, edst, a_s1, a_d1, m1, den1, h, out1);
    k_bias_act<<<GRID1(nNC), T, 0, stream>>>(out1, b1, 1, nNC);   // +b1, ReLU

    // ---------------- layer 2: 128 -> 1 head x 128 ------------------------
    k_gemm_wmma<<<N_NODES / 16, 256, 0, stream>>>(out1, W2, h);   // h2 reuses h
    k_attn<1, OUT_C><<<GRID1(N_NODES), T, 0, stream>>>(
        h, att_src2, att_dst2, a_s2, a_d2);
    k_edge_max<1><<<GRID1(nE2), T, 0, stream>>>(esrc, edst, a_s2, a_d2, m2);
    k_edge_sum<1><<<GRID1(nE2), T, 0, stream>>>(esrc, edst, a_s2, a_d2, m2, den2);
    k_edge_scatter<1, OUT_C><<<GRID1(nE2), T, 0, stream>>>(
        esrc, edst, a_s2, a_d2, m2, den2, h, out);
    k_bias_act<<<GRID1(nNC), T, 0, stream>>>(out, b2, 0, nNC);    // +b2
}